// _GraphSAGE_43671227466095
// MI455X (gfx1250) — compile-verified
//
#include <hip/hip_runtime.h>

typedef __attribute__((ext_vector_type(2))) float v2f;
typedef __attribute__((ext_vector_type(8))) float v8f;

#define N_NODES 100000
#define N_EDGES 1600000
#define FEAT    128      // D_IN == D_HID
#define D_OUT   64
#define MTILE   32       // node rows per GEMM block

// ---------------------------------------------------------------------------
// Edge-degree count (float so we can atomicAdd and divide later)
// ---------------------------------------------------------------------------
__global__ __launch_bounds__(256) void k_count(const long long* __restrict__ dst,
                                               float* __restrict__ cnt) {
  int e = blockIdx.x * 256 + threadIdx.x;          // grid sized exactly E/256
  atomicAdd(&cnt[dst[e]], 1.0f);
}

__global__ __launch_bounds__(256) void k_inv_cnt(float* __restrict__ cnt) {
  int i = blockIdx.x * 256 + threadIdx.x;
  if (i < N_NODES) cnt[i] = 1.0f / fmaxf(cnt[i], 1.0f);
}

// ---------------------------------------------------------------------------
// Scatter-add of 128-float rows: one wave32 per edge, float4 per lane.
// x row (512B) is one coalesced 32-lane b128 load; 4 f32 atomics per lane.
// Node tables fit in 192MB L2, so both sides run at L2 bandwidth.
// ---------------------------------------------------------------------------
__global__ __launch_bounds__(256) void k_scatter(const float* __restrict__ X,
                                                 const long long* __restrict__ src,
                                                 const long long* __restrict__ dst,
                                                 float* __restrict__ agg) {
  int e    = blockIdx.x * 8 + (threadIdx.x >> 5); // grid sized exactly E/8
  int lane = threadIdx.x & 31;
  long long s = src[e];
  long long d = dst[e];
  float4 v = *(const float4*)(X + (size_t)s * FEAT + lane * 4);
  float* a = agg + (size_t)d * FEAT + lane * 4;
  atomicAdd(a + 0, v.x);
  atomicAdd(a + 1, v.y);
  atomicAdd(a + 2, v.z);
  atomicAdd(a + 3, v.w);
}

// ---------------------------------------------------------------------------
// Pack weights into WMMA B-fragment order.
// Logical B[k][n] = (k < K1 ? W1[n][k] : W2[n][k-K1])   (i.e. [W1^T ; W2^T])
// Fragment layout (mirrors documented f32 16x4 A layout):
//   Bp[(((k0/4)*colTiles + ct)*32 + lane)*2 + c] = B[k0 + 2*(lane>>4) + c][ct*16 + (lane&15)]
// Hot loop then reads one coalesced float2 per lane (256B/wave) per k-step.
// ---------------------------------------------------------------------------
__global__ __launch_bounds__(256) void k_pack(const float* __restrict__ W1,
                                              const float* __restrict__ W2,
                                              float* __restrict__ Bp,
                                              int Nout, int K1, int K2) {
  int Ktot = K1 + K2;
  int gid  = blockIdx.x * 256 + threadIdx.x;
  if (gid >= Nout * Ktot) return;
  int c    = gid & 1;
  int l    = (gid >> 1) & 31;
  int rest = gid >> 6;
  int colTiles = Nout >> 4;
  int ct = rest % colTiles;
  int s  = rest / colTiles;
  int k  = s * 4 + 2 * (l >> 4) + c;
  int n  = ct * 16 + (l & 15);
  Bp[gid] = (k < K1) ? W1[(size_t)n * K1 + k] : W2[(size_t)n * K2 + (k - K1)];
}

// ---------------------------------------------------------------------------
// Fused SAGE GEMM:  out = act( [A1*rowScale | A2] @ Bpack + bias )
//   A1: [N, K1] (agg; scaled per-row by rowScale -> mean), A2: [N, K2] or null
//   Bpack: fragment-packed [W1^T ; W2^T], bias: [Nout]
// Block = 32 node rows x Nout cols; 4 waves; each wave owns col-tiles
// ct = wave, wave+4, ... and both 16-row tiles (shares the B fragment).
// fp32 WMMA 16x16x4, K accumulated in 4-wide steps from LDS (padded stride).
// ---------------------------------------------------------------------------
__global__ __launch_bounds__(128) void k_gemm(const float* __restrict__ A1,
                                              const float* __restrict__ rowScale,
                                              const float* __restrict__ A2,
                                              const float* __restrict__ Bp,
                                              const float* __restrict__ bias,
                                              float* __restrict__ out,
                                              int K1, int K2, int Nout, int relu) {
  extern __shared__ float sA[];                 // MTILE x (Ktot+2)
  const int Ktot = K1 + K2;
  const int ldk  = Ktot + 2;                    // +2 floats: kills LDS bank conflicts
  const int rowBase = blockIdx.x * MTILE;
  const int tid = threadIdx.x;

  // Stage A1 (scaled) into LDS columns [0, K1)
  for (int i = tid * 4; i < MTILE * K1; i += 128 * 4) {
    int row = i / K1, col = i - row * K1;
    float4 v = *(const float4*)(A1 + (size_t)(rowBase + row) * K1 + col);
    float sc = rowScale ? rowScale[rowBase + row] : 1.0f;
    float* sp = sA + row * ldk + col;
    sp[0] = v.x * sc; sp[1] = v.y * sc; sp[2] = v.z * sc; sp[3] = v.w * sc;
  }
  // Stage A2 into LDS columns [K1, K1+K2)
  if (A2) {
    for (int i = tid * 4; i < MTILE * K2; i += 128 * 4) {
      int row = i / K2, col = i - row * K2;
      float4 v = *(const float4*)(A2 + (size_t)(rowBase + row) * K2 + col);
      float* sp = sA + row * ldk + K1 + col;
      sp[0] = v.x; sp[1] = v.y; sp[2] = v.z; sp[3] = v.w;
    }
  }
  __syncthreads();

  const int wave = tid >> 5, lane = tid & 31;
  const int pl = lane & 15, h = lane >> 4;      // A/B fragment coords
  const int colTiles = Nout >> 4;
  const int ksteps   = Ktot >> 2;
  const float* a0base = sA + pl * ldk;          // row tile 0, M = pl
  const float* a1base = sA + (16 + pl) * ldk;   // row tile 1, M = 16 + pl

  for (int ct = wave; ct < colTiles; ct += 4) {
    v8f acc0 = {}; v8f acc1 = {};
    const v2f* bptr = (const v2f*)Bp + (size_t)ct * 32 + lane;
    const size_t bstride = (size_t)colTiles * 32;
    for (int s = 0; s < ksteps; ++s) {
      v2f b  = bptr[(size_t)s * bstride];                 // coalesced, L2-resident
      int  k = s * 4 + 2 * h;
      v2f a0 = *(const v2f*)(a0base + k);                 // ds_load_b64, conflict-free
      v2f a1 = *(const v2f*)(a1base + k);
      acc0 = __builtin_amdgcn_wmma_f32_16x16x4_f32(false, a0, false, b, (short)0,
                                                   acc0, false, false);
      acc1 = __builtin_amdgcn_wmma_f32_16x16x4_f32(false, a1, false, b, (short)0,
                                                   acc1, false, false);
    }
    // C/D layout: VGPR r holds M = r + 8*h (lanes 0-15 / 16-31), N = pl
    float bc = bias[ct * 16 + pl];
    #pragma unroll
    for (int r = 0; r < 8; ++r) {
      float v0 = acc0[r] + bc;
      float v1 = acc1[r] + bc;
      if (relu) { v0 = fmaxf(v0, 0.0f); v1 = fmaxf(v1, 0.0f); }
      out[(size_t)(rowBase + r + 8 * h) * Nout + ct * 16 + pl]      = v0;
      out[(size_t)(rowBase + 16 + r + 8 * h) * Nout + ct * 16 + pl] = v1;
    }
  }
}

// ---------------------------------------------------------------------------
extern "C" void kernel_launch(void* const* d_in, const int* in_sizes, int n_in,
                              void* d_out, int out_size, void* d_ws, size_t ws_size,
                              hipStream_t stream) {
  const float*     x    = (const float*)d_in[0];
  const long long* edge = (const long long*)d_in[1];
  const float*     W1l  = (const float*)d_in[2];
  const float*     b1l  = (const float*)d_in[3];
  const float*     W1r  = (const float*)d_in[4];
  const float*     W2l  = (const float*)d_in[5];
  const float*     b2l  = (const float*)d_in[6];
  const float*     W2r  = (const float*)d_in[7];
  const float*     Wh   = (const float*)d_in[8];
  const float*     bh   = (const float*)d_in[9];
  const long long* src  = edge;
  const long long* dst  = edge + N_EDGES;
  float* out = (float*)d_out;

  // Workspace carve-up (~154 MB)
  float* agg = (float*)d_ws;                         // N*128
  float* cnt = agg + (size_t)N_NODES * FEAT;         // N (padded)
  float* h1  = cnt + 100096;                         // N*128
  float* h2  = h1 + (size_t)N_NODES * FEAT;          // N*128
  float* Bp1 = h2 + (size_t)N_NODES * FEAT;          // 256*128
  float* Bp2 = Bp1 + 256 * FEAT;                     // 256*128
  float* BpH = Bp2 + 256 * FEAT;                     // 128*64

  const int smemL = MTILE * (2 * FEAT + 2) * sizeof(float); // 33 KB (Ktot=256)
  const int smemH = MTILE * (FEAT + 2) * sizeof(float);     // 16.3 KB (Ktot=128)

  // Degree counts (shared by both layers)
  hipMemsetAsync(cnt, 0, N_NODES * sizeof(float), stream);
  k_count<<<N_EDGES / 256, 256, 0, stream>>>(dst, cnt);
  k_inv_cnt<<<(N_NODES + 255) / 256, 256, 0, stream>>>(cnt);

  // One-time weight fragment packing (weights are tiny, stays in L2)
  k_pack<<<(FEAT * 256 + 255) / 256, 256, 0, stream>>>(W1l, W1r, Bp1, FEAT, FEAT, FEAT);
  k_pack<<<(FEAT * 256 + 255) / 256, 256, 0, stream>>>(W2l, W2r, Bp2, FEAT, FEAT, FEAT);
  k_pack<<<(D_OUT * FEAT + 255) / 256, 256, 0, stream>>>(Wh, nullptr, BpH, D_OUT, FEAT, 0);

  // Layer 1: scatter-mean + fused dual GEMM + bias + ReLU
  hipMemsetAsync(agg, 0, (size_t)N_NODES * FEAT * sizeof(float), stream);
  k_scatter<<<N_EDGES / 8, 256, 0, stream>>>(x, src, dst, agg);
  k_gemm<<<N_NODES / MTILE, 128, smemL, stream>>>(agg, cnt, x, Bp1, b1l, h1,
                                                  FEAT, FEAT, FEAT, 1);

  // Layer 2
  hipMemsetAsync(agg, 0, (size_t)N_NODES * FEAT * sizeof(float), stream);
  k_scatter<<<N_EDGES / 8, 256, 0, stream>>>(h1, src, dst, agg);
  k_gemm<<<N_NODES / MTILE, 128, smemL, stream>>>(agg, cnt, h1, Bp2, b2l, h2,
                                                  FEAT, FEAT, FEAT, 1);

  // Head: h2 @ Wh^T + bh  (no relu)
  k_gemm<<<N_NODES / MTILE, 128, smemH, stream>>>(h2, nullptr, nullptr, BpH, bh, out,
                                                  FEAT, 0, D_OUT, 0);
}